// AdaptiveSkeletonLoss_78967268704620
// MI455X (gfx1250) — compile-verified
//
#include <hip/hip_runtime.h>
#include <stdint.h>

#define B_      32
#define H_      512
#define W_      512
#define WPR     16            // 32-bit mask words per 512-px row
#define RROWS   64            // central rows per stripe
#define HALO    10
#define SROWS   (RROWS + 2*HALO)   // 84
#define NSTRIPE (H_ / RROWS)       // 8
#define NCHUNK  (SROWS / 4)        // 21

// ---- CDNA5 async global->LDS path (gfx1250) ---------------------------------
__device__ __forceinline__ void async_copy16(uint32_t lds_byte, const void* gaddr) {
  asm volatile("global_load_async_to_lds_b128 %0, %1, off"
               :: "v"(lds_byte), "v"(gaddr) : "memory");
}
__device__ __forceinline__ void wait_async0() { asm volatile("s_wait_asynccnt 0" ::: "memory"); }
__device__ __forceinline__ void wait_async3() { asm volatile("s_wait_asynccnt 3" ::: "memory"); }
__device__ __forceinline__ void wait_async4() { asm volatile("s_wait_asynccnt 4" ::: "memory"); }

__device__ __forceinline__ uint32_t ballot32(bool p) { return __builtin_amdgcn_ballot_w32(p); }

__global__ __launch_bounds__(256, 2)
void skel_main(const float* __restrict__ pred, const float* __restrict__ gt,
               uint32_t* __restrict__ ws) {
  __shared__ float    fbuf[8192];            // 32 KB float staging (async target)
  __shared__ uint32_t ppack[SROWS*WPR];      // pred bit mask, 84x16
  __shared__ uint32_t gpack[SROWS*WPR];      // gt bit mask
  __shared__ uint32_t curb[SROWS*WPR];       // dilation working buffer
  __shared__ uint32_t hbuf[SROWS*WPR];       // horizontal-dilation buffer
  __shared__ float    fred[8];
  __shared__ uint32_t ured[8];

  const int tid  = threadIdx.x;
  const int lane = tid & 31;
  const int wv   = tid >> 5;
  const int blk  = blockIdx.x;
  const int b    = blk >> 3;
  const int s    = blk & 7;
  const int r0   = s * RROWS;

  const size_t imgbase = (size_t)b * (H_ * W_);
  const float* pimg = pred + imgbase;
  const float* gimg = gt + imgbase;
  const uint32_t ldsF = (uint32_t)(uintptr_t)(&fbuf[0]);

  // ---------------- Phase 1: stream + binarize (bit-pack via wave32 ballot) --
  auto load_chunk = [&](int c, int buf) {   // 4 rows x 2 images, 4 async/thread
    const int rowbase = r0 - HALO + c * 4;
#pragma unroll
    for (int k = 0; k < 2; ++k) {
      int u  = tid + k * 256;               // 0..511 units of 16B over 4 rows
      int rq = u >> 7;
      int c4 = u & 127;
      int gr  = rowbase + rq;
      int grc = gr < 0 ? 0 : (gr > H_ - 1 ? H_ - 1 : gr);
      size_t off = (size_t)grc * W_ + (size_t)(c4 * 4);
      uint32_t l0 = ldsF + (uint32_t)(((buf * 2 + 0) * 2048 + u * 4) * 4);
      uint32_t l1 = ldsF + (uint32_t)(((buf * 2 + 1) * 2048 + u * 4) * 4);
      async_copy16(l0, pimg + off);
      async_copy16(l1, gimg + off);
    }
  };

  float sp = 0.f, sg = 0.f, spg = 0.f;      // dice partials (raw floats)

  auto pack_chunk = [&](int c, int buf) {
    for (int q = wv; q < 64; q += 8) {      // (row-in-chunk, word) pairs
      int rq = q >> 4, wq = q & 15;
      int srow = c * 4 + rq;
      int gr = r0 - HALO + srow;
      int idx = rq * 512 + wq * 32 + lane;
      float pv = fbuf[(buf * 2 + 0) * 2048 + idx];
      float gv = fbuf[(buf * 2 + 1) * 2048 + idx];
      uint32_t pmw = ballot32(pv > 0.5f);
      uint32_t gmw = ballot32(gv > 0.5f);
      if (gr < 0 || gr >= H_) { pmw = 0u; gmw = 0u; }   // zero padding rows
      if (lane == 0) { ppack[srow * WPR + wq] = pmw; gpack[srow * WPR + wq] = gmw; }
      if (gr >= r0 && gr < r0 + RROWS) { sp += pv; sg += gv; spg += pv * gv; }
    }
  };

  load_chunk(0, 0);
  for (int c = 0; c < NCHUNK; ++c) {
    if (c + 1 < NCHUNK) { load_chunk(c + 1, (c + 1) & 1); wait_async4(); }
    else wait_async0();
    __syncthreads();
    pack_chunk(c, c & 1);
    __syncthreads();
  }

  // ---------------- Phase 2 prefetch (hidden behind medial passes) -----------
  auto load_group = [&](int g, int buf) {   // 6 pred rows, 3 async/thread
    const int grp0 = r0 + g * 4;
#pragma unroll
    for (int k = 0; k < 3; ++k) {
      int u  = tid + k * 256;               // 0..767
      int j  = u >> 7;                      // 0..5
      int c4 = u & 127;
      int gr  = grp0 - 1 + j;
      int grc = gr < 0 ? 0 : (gr > H_ - 1 ? H_ - 1 : gr);
      size_t off = (size_t)grc * W_ + (size_t)(c4 * 4);
      uint32_t lo = ldsF + (uint32_t)((buf * 3072 + j * 512 + c4 * 4) * 4);
      async_copy16(lo, pimg + off);
    }
  };
  load_group(0, 0);

  // ---------------- Medial: 10 binary dilations on bitboards -----------------
  auto medial = [&](const uint32_t* __restrict__ ref, const uint32_t* __restrict__ tgt,
                    uint32_t& dsum, uint32_t& pts) {
    __syncthreads();
    for (int j = tid; j < SROWS * WPR; j += 256) curb[j] = ref[j];
    uint32_t tg[4], fnd[4];
#pragma unroll
    for (int i = 0; i < 4; ++i) {           // 4 owned central words per thread
      int q = tid * 4 + i;
      int rc = q >> 4, w = q & 15;
      tg[i]  = tgt[(rc + HALO) * WPR + w];
      fnd[i] = 0u;
    }
    dsum = 0u;
    for (int d = 1; d <= 10; ++d) {
      __syncthreads();
      for (int j = tid; j < SROWS * WPR; j += 256) {   // horizontal OR-shift
        int w = j & 15;
        uint32_t m = curb[j];
        uint32_t L = w ? curb[j - 1] : 0u;
        uint32_t R = (w < 15) ? curb[j + 1] : 0u;
        hbuf[j] = m | (m << 1) | (m >> 1) | (L >> 31) | (R << 31);
      }
      __syncthreads();
      for (int j = tid; j < SROWS * WPR; j += 256) {   // vertical OR
        uint32_t v = hbuf[j];
        if (j >= WPR) v |= hbuf[j - WPR];
        if (j < (SROWS - 1) * WPR) v |= hbuf[j + WPR];
        curb[j] = v;
      }
      __syncthreads();
#pragma unroll
      for (int i = 0; i < 4; ++i) {
        int q = tid * 4 + i;
        int rc = q >> 4, w = q & 15;
        uint32_t nw = curb[(rc + HALO) * WPR + w] & tg[i] & ~fnd[i];
        fnd[i] |= nw;
        dsum += (uint32_t)d * (uint32_t)__popc(nw);
      }
    }
    pts = 0u;
#pragma unroll
    for (int i = 0; i < 4; ++i) {
      dsum += 10u * (uint32_t)__popc(tg[i] & ~fnd[i]);  // unfound -> MAX_DIST
      pts  += (uint32_t)__popc(tg[i]);
    }
  };

  uint32_t p2g_d, p2g_p, g2p_d, g2p_p;
  medial(gpack, ppack, p2g_d, p2g_p);       // dilate gt, target pred
  medial(ppack, gpack, g2p_d, g2p_p);       // dilate pred, target gt

  // ---------------- Phase 2: structural classification -----------------------
  uint32_t cie = 0, cpe = 0, cge = 0, cim = 0, cpm = 0, cgm = 0, cij = 0, cpj = 0, cgj = 0;
  for (int g = 0; g < RROWS / 4; ++g) {     // 16 groups of 4 central rows
    const int buf = g & 1;
    if (g + 1 < RROWS / 4) { load_group(g + 1, (g + 1) & 1); wait_async3(); }
    else wait_async0();
    __syncthreads();
    const int grp0 = r0 + g * 4;
#pragma unroll
    for (int j = 0; j < 6; ++j) {           // zero out-of-image rows
      int gr = grp0 - 1 + j;
      if (gr < 0 || gr >= H_) {
        for (int cc = tid; cc < 512; cc += 256) fbuf[buf * 3072 + j * 512 + cc] = 0.f;
      }
    }
    __syncthreads();
    for (int q = wv; q < 64; q += 8) {
      int rq = q >> 4, w = q & 15;
      int gr = grp0 + rq;
      int cc = w * 32 + lane;
      const float* fb = &fbuf[buf * 3072];
      float Am1 = (cc > 0)   ? fb[(rq    ) * 512 + cc - 1] : 0.f;
      float A0  =              fb[(rq    ) * 512 + cc];
      float Ap1 = (cc < 511) ? fb[(rq    ) * 512 + cc + 1] : 0.f;
      float Bm1 = (cc > 0)   ? fb[(rq + 1) * 512 + cc - 1] : 0.f;
      float B0  =              fb[(rq + 1) * 512 + cc];
      float Bp1 = (cc < 511) ? fb[(rq + 1) * 512 + cc + 1] : 0.f;
      float Cm1 = (cc > 0)   ? fb[(rq + 2) * 512 + cc - 1] : 0.f;
      float C0  =              fb[(rq + 2) * 512 + cc];
      float Cp1 = (cc < 511) ? fb[(rq + 2) * 512 + cc + 1] : 0.f;
      float n = Am1 + A0 + Ap1 + Bm1 + Bp1 + Cm1 + C0 + Cp1;   // raw conv (pred)
      bool onp = B0 > 0.5f;
      uint32_t pew = ballot32(onp && (n == 1.0f));
      uint32_t pmw = ballot32(onp && (n == 2.0f));
      uint32_t pjw = ballot32(onp && (n >  2.0f));
      // gt is exactly 0/1: exact neighbor counts via CSA popcount tree on bitplanes
      int si = gr - (r0 - HALO);
      uint32_t ge_w, gm_w, gj_w;
      {
        const int rm = (si - 1) * WPR, rc = si * WPR, rp = (si + 1) * WPR;
        uint32_t aM = gpack[rm + w], aL = w ? gpack[rm + w - 1] : 0u, aR = (w < 15) ? gpack[rm + w + 1] : 0u;
        uint32_t bM = gpack[rc + w], bL = w ? gpack[rc + w - 1] : 0u, bR = (w < 15) ? gpack[rc + w + 1] : 0u;
        uint32_t cM = gpack[rp + w], cL = w ? gpack[rp + w - 1] : 0u, cR = (w < 15) ? gpack[rp + w + 1] : 0u;
        uint32_t al = (aM << 1) | (aL >> 31), ar = (aM >> 1) | (aR << 31);
        uint32_t bl = (bM << 1) | (bL >> 31), br = (bM >> 1) | (bR << 31);
        uint32_t cl = (cM << 1) | (cL >> 31), cr = (cM >> 1) | (cR << 31);
        uint32_t t, s1a, c1a, s1b, c1b, s1c, c1c;
        t = al ^ aM;   s1a = t ^ ar;   c1a = (al & aM) | (ar & t);
        t = bl ^ br;   s1b = t ^ cl;   c1b = (bl & br) | (cl & t);
        t = cM ^ cr;   s1c = t ^ s1a;  c1c = (cM & cr) | (s1a & t);
        uint32_t s0 = s1b ^ s1c, c1d = s1b & s1c;
        t = c1a ^ c1b; uint32_t s2a = t ^ c1c, c2a = (c1a & c1b) | (c1c & t);
        uint32_t s1 = s2a ^ c1d, c2b = s2a & c1d;
        uint32_t s2 = c2a ^ c2b, s3 = c2a & c2b;     // n = s0 + 2s1 + 4s2 + 8s3
        uint32_t on = bM;
        ge_w = on &  s0 & ~s1 & ~s2 & ~s3;           // n == 1
        gm_w = on & ~s0 &  s1 & ~s2 & ~s3;           // n == 2
        gj_w = on & ((s0 & s1) | s2 | s3);           // n  > 2
      }
      if (lane == 0) {
        cie += __popc(pew & ge_w); cpe += __popc(pew); cge += __popc(ge_w);
        cim += __popc(pmw & gm_w); cpm += __popc(pmw); cgm += __popc(gm_w);
        cij += __popc(pjw & gj_w); cpj += __popc(pjw); cgj += __popc(gj_w);
      }
    }
    __syncthreads();
  }

  // ---------------- Deterministic block reductions -> workspace slot ---------
  auto redF = [&](float v) -> float {
#pragma unroll
    for (int o = 16; o > 0; o >>= 1) v += __shfl_down(v, o, 32);
    if (lane == 0) fred[wv] = v;
    __syncthreads();
    float t = 0.f;
    if (tid == 0) for (int i = 0; i < 8; ++i) t += fred[i];
    __syncthreads();
    return t;
  };
  auto redU = [&](uint32_t v) -> uint32_t {
#pragma unroll
    for (int o = 16; o > 0; o >>= 1) v += __shfl_down(v, o, 32);
    if (lane == 0) ured[wv] = v;
    __syncthreads();
    uint32_t t = 0u;
    if (tid == 0) for (int i = 0; i < 8; ++i) t += ured[i];
    __syncthreads();
    return t;
  };

  uint32_t* slot = ws + (size_t)blk * 16;
  float rf;
  rf = redF(spg); if (tid == 0) slot[0] = __float_as_uint(rf);
  rf = redF(sp);  if (tid == 0) slot[1] = __float_as_uint(rf);
  rf = redF(sg);  if (tid == 0) slot[2] = __float_as_uint(rf);
  uint32_t rv;
  rv = redU(cie);   if (tid == 0) slot[3]  = rv;
  rv = redU(cpe);   if (tid == 0) slot[4]  = rv;
  rv = redU(cge);   if (tid == 0) slot[5]  = rv;
  rv = redU(cim);   if (tid == 0) slot[6]  = rv;
  rv = redU(cpm);   if (tid == 0) slot[7]  = rv;
  rv = redU(cgm);   if (tid == 0) slot[8]  = rv;
  rv = redU(cij);   if (tid == 0) slot[9]  = rv;
  rv = redU(cpj);   if (tid == 0) slot[10] = rv;
  rv = redU(cgj);   if (tid == 0) slot[11] = rv;
  rv = redU(p2g_d); if (tid == 0) slot[12] = rv;
  rv = redU(p2g_p); if (tid == 0) slot[13] = rv;
  rv = redU(g2p_d); if (tid == 0) slot[14] = rv;
  rv = redU(g2p_p); if (tid == 0) slot[15] = rv;
}

// ---------------- Finisher: per-batch losses + balanced combine --------------
__global__ void skel_final(const uint32_t* __restrict__ ws, float* __restrict__ out) {
  const int b = threadIdx.x;                 // 32 threads = 1 wave32
  float inter = 0.f, ps = 0.f, gs = 0.f;
  uint32_t icnt[13];
  for (int k = 0; k < 13; ++k) icnt[k] = 0u;
  for (int s = 0; s < NSTRIPE; ++s) {
    const uint32_t* p = ws + (size_t)(b * NSTRIPE + s) * 16;
    inter += __uint_as_float(p[0]);
    ps    += __uint_as_float(p[1]);
    gs    += __uint_as_float(p[2]);
    for (int k = 0; k < 13; ++k) icnt[k] += p[3 + k];
  }
  float cnt[13];
  for (int k = 0; k < 13; ++k) cnt[k] = (float)icnt[k];
  float dice = (2.f * inter + 1.f) / (ps + gs + 1.f);
  float ie = cnt[0], pe = cnt[1], ge = cnt[2];
  float im = cnt[3], pm = cnt[4], gm = cnt[5];
  float ij = cnt[6], pj = cnt[7], gj = cnt[8];
  float eiou = (ie + 1.f) / (pe + ge - ie + 1.f);
  float miou = (im + 1.f) / (pm + gm - im + 1.f);
  float jiou = (ij + 1.f) / (pj + gj - ij + 1.f);
  float total = ge + gj + gm + 1.f;
  float sloss = 1.f - ((ge / total) * eiou + (gj / total) * jiou + (gm / total) * miou);
  float p2g = cnt[9]  / (cnt[10] + 1.f);
  float g2p = cnt[11] / (cnt[12] + 1.f);
  float med = ((p2g + g2p) * 0.5f) * 0.1f;   // / MAX_DIST
  float dsum = dice, ssum = sloss, msum = med;
  for (int o = 16; o > 0; o >>= 1) {
    dsum += __shfl_down(dsum, o, 32);
    ssum += __shfl_down(ssum, o, 32);
    msum += __shfl_down(msum, o, 32);
  }
  if (b == 0) {
    float dice_loss  = 1.f - dsum * (1.f / 32.f);
    float structural = ssum * (1.f / 32.f);
    float medial     = msum * (1.f / 32.f);
    float avg = (dice_loss + structural + medial) * (1.f / 3.f);
    out[0] = dice_loss  / (dice_loss  + 1.f) * avg
           + structural / (structural + 1.f) * avg
           + medial     / (medial     + 1.f) * avg;
  }
}

extern "C" void kernel_launch(void* const* d_in, const int* in_sizes, int n_in,
                              void* d_out, int out_size, void* d_ws, size_t ws_size,
                              hipStream_t stream) {
  (void)in_sizes; (void)n_in; (void)out_size; (void)ws_size;
  const float* pred = (const float*)d_in[0];
  const float* gtp  = (const float*)d_in[1];
  uint32_t* ws = (uint32_t*)d_ws;     // 256 slots x 16 u32 = 16 KB
  float* out = (float*)d_out;
  skel_main<<<dim3(B_ * NSTRIPE), dim3(256), 0, stream>>>(pred, gtp, ws);
  skel_final<<<dim3(1), dim3(32), 0, stream>>>(ws, out);
}